// AffineLayer2d_5188320493995
// MI455X (gfx1250) — compile-verified
//
#include <hip/hip_runtime.h>
#include <hip/hip_bf16.h>

typedef __attribute__((ext_vector_type(2))) float v2f;
typedef __attribute__((ext_vector_type(8))) float v8f;

#define PI_F 3.14159265358979323846f

// Problem constants (from reference setup_inputs)
#define N_  8
#define S_  32
#define C_  3
#define H_  224
#define W_  224
#define WT_ (W_ / 16)          // 14 pixel tiles per row
#define SG_ 4                  // 4 sample groups of 8
// waves total = N_ * SG_ * H_ * WT_ = 100352 ; blocks = /8 = 12544
#define BLOCKS_ ((N_ * SG_ * H_ * WT_) / 8)

// ---------------------------------------------------------------------------
// Stage 1: 256 threads, one per (n,s): build M = sum_k coeff[k]*k'[k]*G[k],
// expm via scaling (2^-6) + 12-term Taylor + 6 squarings. Store rows 0,1
// (theta[b][2][3]) to workspace.
// ---------------------------------------------------------------------------
__device__ __forceinline__ void mm3(const float* a, const float* b, float* c) {
#pragma unroll
    for (int i = 0; i < 3; ++i)
#pragma unroll
        for (int j = 0; j < 3; ++j)
            c[i * 3 + j] = fmaf(a[i * 3 + 0], b[0 * 3 + j],
                           fmaf(a[i * 3 + 1], b[1 * 3 + j],
                                a[i * 3 + 2] * b[2 * 3 + j]));
}

__global__ __launch_bounds__(256) void expm_theta_kernel(
        const float* __restrict__ ksamp,   // [6, N, S]
        const float* __restrict__ rf,      // [6]
        float* __restrict__ theta) {       // [N*S, 2, 3]
    const int t = threadIdx.x;             // 0..255
    const int n = t >> 5;                  // S_ == 32
    const int s = t & 31;

    float c0 = rf[0], c1 = rf[1], c2 = rf[2], c3 = rf[3], c4 = rf[4], c5 = rf[5];
    c2 = fminf(fmaxf(c2, -PI_F), PI_F);

    const int NS = N_ * S_;
    const int base = n * S_ + s;
    const float k0 = fmaf(ksamp[0 * NS + base], 2.0f, -1.0f);
    const float k1 = fmaf(ksamp[1 * NS + base], 2.0f, -1.0f);
    const float k2 = fmaf(ksamp[2 * NS + base], 2.0f, -1.0f);
    const float k3 = fmaf(ksamp[3 * NS + base], 2.0f, -1.0f);
    const float k4 = fmaf(ksamp[4 * NS + base], 2.0f, -1.0f);
    const float k5 = fmaf(ksamp[5 * NS + base], 2.0f, -1.0f);

    // M from generators: G1:(0,2) G2:(1,2) G3:(0,1)=-1,(1,0)=+1 G4:(0,0) G5:(1,1) G6:(0,1),(1,0)
    float A[9];
    A[0] = c3 * k3;              A[1] = c5 * k5 - c2 * k2;  A[2] = c0 * k0;
    A[3] = c2 * k2 + c5 * k5;    A[4] = c4 * k4;            A[5] = c1 * k1;
    A[6] = 0.f;                  A[7] = 0.f;                A[8] = 0.f;
#pragma unroll
    for (int i = 0; i < 9; ++i) A[i] *= (1.0f / 64.0f);   // 2^-6 scaling

    float out[9] = {1.f + A[0], A[1], A[2],
                    A[3], 1.f + A[4], A[5],
                    A[6], A[7], 1.f + A[8]};
    float term[9];
#pragma unroll
    for (int i = 0; i < 9; ++i) term[i] = A[i];

#pragma unroll
    for (int it = 2; it <= 12; ++it) {
        float nt[9];
        mm3(term, A, nt);
        const float inv = 1.0f / (float)it;
#pragma unroll
        for (int i = 0; i < 9; ++i) { term[i] = nt[i] * inv; out[i] += term[i]; }
    }
#pragma unroll
    for (int sq = 0; sq < 6; ++sq) {
        float nt[9];
        mm3(out, out, nt);
#pragma unroll
        for (int i = 0; i < 9; ++i) out[i] = nt[i];
    }

    float* th = theta + base * 6;
#pragma unroll
    for (int i = 0; i < 6; ++i) th[i] = out[i];
}

// ---------------------------------------------------------------------------
// Stage 2: fused affine_grid + bilinear grid_sample.
// One wave per (n, sample-group-of-8, h, 16-pixel w-tile).
// One V_WMMA_F32_16X16X4_F32 per wave computes the 16x16 coordinate block:
//   A (16x4): rows 2j / 2j+1 = [t00 t01 t02 0] / [t10 t11 t12 0] of sample j
//   B (4x16): columns = [gx; gy; 1; 0] for 16 consecutive pixels
//   D row 2j = ix_norm, row 2j+1 = iy_norm for sample j, all 16 pixels.
// Each lane then does bilinear sampling (L2-resident source) for 4 samples x
// 3 channels = 12 outputs, stored nontemporally (output is write-once 154MB).
// ---------------------------------------------------------------------------
__global__ __launch_bounds__(256) void affine_sample_kernel(
        const float* __restrict__ x,       // [N, C, H, W]
        const float* __restrict__ theta,   // [N*S, 2, 3]
        float* __restrict__ out) {         // [N, S, C, H, W]
    const int lane = threadIdx.x & 31;
    const int wave = threadIdx.x >> 5;
    int gw = blockIdx.x * 8 + wave;        // 0 .. 100351, exact cover

    const int wt = gw % WT_;  gw /= WT_;
    const int h  = gw % H_;   gw /= H_;
    const int sg = gw & 3;
    const int n  = gw >> 2;

    const int half = lane >> 4;            // 0: K0/K1 half, 1: K2/K3 half
    const int col  = lane & 15;            // pixel column within tile / A row

    // ---- A matrix: 16x4 f32. M = lane%16; lanes 0-15 hold K0,K1; 16-31 hold K2,K3.
    const int m      = lane & 15;
    const int s_row  = m >> 1;             // sample 0..7 within group
    const int comp   = m & 1;              // 0 = x-row, 1 = y-row
    const float* th  = theta + ((n * S_ + sg * 8 + s_row) * 2 + comp) * 3;
    const float t0 = th[0], t1 = th[1], t2 = th[2];
    v2f a;
    a.x = half ? t2 : t0;
    a.y = half ? 0.0f : t1;

    // ---- B matrix: 4x16 f32. N = lane%16; lanes 0-15 hold K0,K1; 16-31 hold K2,K3.
    const int w = wt * 16 + col;
    const float gx = fmaf((float)w, 2.0f / (float)(W_ - 1), -1.0f);
    const float gy = fmaf((float)h, 2.0f / (float)(H_ - 1), -1.0f);
    v2f b;
    b.x = half ? 1.0f : gx;
    b.y = half ? 0.0f : gy;

    v8f cz = {};
    // D = A x B : emits v_wmma_f32_16x16x4_f32
    v8f d = __builtin_amdgcn_wmma_f32_16x16x4_f32(
        /*neg_a=*/false, a, /*neg_b=*/false, b,
        /*c_mod=*/(short)0, cz, /*reuse_a=*/false, /*reuse_b=*/false);

    // Lane l (half hh, col c): D[r] = row (r + 8*hh), col c
    //  -> sample (hh*4 + r/2), component r%2.
    const int sBase = sg * 8 + half * 4;   // sample index within S for j=0

#pragma unroll
    for (int j = 0; j < 4; ++j) {
        const float ixn = d[2 * j];
        const float iyn = d[2 * j + 1];
        // align_corners=True mapping
        const float ix = (ixn + 1.0f) * (0.5f * (float)(W_ - 1));
        const float iy = (iyn + 1.0f) * (0.5f * (float)(H_ - 1));
        const float x0f = floorf(ix);
        const float y0f = floorf(iy);
        const float wx1 = ix - x0f, wx0 = 1.0f - wx1;
        const float wy1 = iy - y0f, wy0 = 1.0f - wy1;
        const int x0 = (int)x0f, y0 = (int)y0f;
        const int x1 = x0 + 1,   y1 = y0 + 1;

        const float vx0 = (x0 >= 0 && x0 <= W_ - 1) ? 1.0f : 0.0f;
        const float vx1 = (x1 >= 0 && x1 <= W_ - 1) ? 1.0f : 0.0f;
        const float vy0 = (y0 >= 0 && y0 <= H_ - 1) ? 1.0f : 0.0f;
        const float vy1 = (y1 >= 0 && y1 <= H_ - 1) ? 1.0f : 0.0f;

        const int cx0 = min(max(x0, 0), W_ - 1);
        const int cx1 = min(max(x1, 0), W_ - 1);
        const int cy0 = min(max(y0, 0), H_ - 1);
        const int cy1 = min(max(y1, 0), H_ - 1);

        const float w00 = wy0 * wx0 * vy0 * vx0;
        const float w01 = wy0 * wx1 * vy0 * vx1;
        const float w10 = wy1 * wx0 * vy1 * vx0;
        const float w11 = wy1 * wx1 * vy1 * vx1;

        const int i00 = cy0 * W_ + cx0;
        const int i01 = cy0 * W_ + cx1;
        const int i10 = cy1 * W_ + cx0;
        const int i11 = cy1 * W_ + cx1;

        const int s_abs = sBase + j;                     // 0..31
        const size_t obase =
            ((((size_t)(n * S_ + s_abs)) * C_) * H_ + h) * W_ + w;

#pragma unroll
        for (int c = 0; c < C_; ++c) {
            const float* plane = x + ((size_t)(n * C_ + c)) * (H_ * W_);
            const float v00 = plane[i00];
            const float v01 = plane[i01];
            const float v10 = plane[i10];
            const float v11 = plane[i11];
            const float r = fmaf(v00, w00, fmaf(v01, w01,
                            fmaf(v10, w10, v11 * w11)));
            __builtin_nontemporal_store(r, &out[obase + (size_t)c * (H_ * W_)]);
        }
    }
}

// ---------------------------------------------------------------------------
extern "C" void kernel_launch(void* const* d_in, const int* in_sizes, int n_in,
                              void* d_out, int out_size, void* d_ws, size_t ws_size,
                              hipStream_t stream) {
    const float* x     = (const float*)d_in[0];   // [8,3,224,224]
    const float* ksamp = (const float*)d_in[1];   // [6,8,32]
    const float* rf    = (const float*)d_in[2];   // [6]
    float* out   = (float*)d_out;                 // [8,32,3,224,224]
    float* theta = (float*)d_ws;                  // [256,2,3] = 6 KB scratch

    expm_theta_kernel<<<1, 256, 0, stream>>>(ksamp, rf, theta);
    affine_sample_kernel<<<BLOCKS_, 256, 0, stream>>>(x, theta, out);
}